// Crf_89489938579846
// MI455X (gfx1250) — compile-verified
//
#include <hip/hip_runtime.h>
#include <stdint.h>

// CRF Viterbi decode, MI455X (gfx1250, wave32).
// B=1024 sequences, T=1024 steps, C=41 tags.
//
// Phase 1 (crf_forward): one wave32 per sequence. Lane l owns tag columns
//   j0=l and (for l<9) j1=l+32. Score vector ping-pongs in LDS; since the
//   workgroup is a single wave, barriers are hardware NOPs and LDS ordering
//   comes from the wave's own DScnt — zero barrier cost across 1024 steps.
//   Emissions are double-buffered in LDS via GLOBAL_LOAD_ASYNC_TO_LDS_B128
//   (ASYNCcnt) overlapped with compute; history stored as uint8 [B][T-1][C].
// Phase 2 (crf_backtrace): one thread per sequence, dependent byte-load chain
//   broken with GLOBAL_PREFETCH_B8 software prefetch.
//
// WMMA intentionally absent: Viterbi is a max-plus (tropical) matrix product
// with argmax history — not expressible as A*B+C on the matrix pipes.
// Roofline: 172 MB emissions + 2x43 MB history ~ 11 us floor @ 23.3 TB/s;
// the binding resource is VALU (B*T*C^2 ~ 1.76G max-add steps).

#define C      41
#define TSTEPS 1024
#define BATCH  1024
#define TT     32                    // time-tile (steps per LDS tile)
#define NT     (TSTEPS / TT)         // 32 tiles
#define CHUNKS ((TT * C * 4) / 16)   // 328 float4 chunks per tile
#define NTH    32                    // ONE wave32 per block

#if __has_builtin(__builtin_amdgcn_global_load_async_to_lds_b128)
#define HAVE_ASYNC 1
#else
#define HAVE_ASYNC 0
#endif

// Async-to-LDS builtin takes addrspace(1)/(3) pointers to a 16-byte vector.
typedef int v4i __attribute__((vector_size(16)));
typedef __attribute__((address_space(1))) v4i GV4;
typedef __attribute__((address_space(3))) v4i LV4;

__device__ __forceinline__ void async_wait0() {
#if __has_builtin(__builtin_amdgcn_s_wait_asynccnt)
    __builtin_amdgcn_s_wait_asynccnt(0);
#elif HAVE_ASYNC
    asm volatile("s_wait_asynccnt 0" ::: "memory");
#endif
}

// Copy one emission tile (CHUNKS x 16B, contiguous) global -> LDS.
__device__ __forceinline__ void copy_tile(const float4* __restrict__ g,
                                          float* lds, int tid) {
    float4* l4 = (float4*)lds;
#if HAVE_ASYNC
    for (int idx = tid; idx < CHUNKS; idx += NTH) {
        __builtin_amdgcn_global_load_async_to_lds_b128(
            (GV4*)(g + idx), (LV4*)(l4 + idx), 0, 0);
    }
#else
    for (int idx = tid; idx < CHUNKS; idx += NTH) l4[idx] = g[idx];
#endif
}

// Max-plus column step: best/arg of sv[i] + tc[i] over i, first-max wins.
// Even/odd split halves the cmp/cndmask dependency chain; the merge prefers
// the smaller index on value ties, reproducing jnp.argmax semantics.
__device__ __forceinline__ void maxplus_col(const float* sv, const float* tc,
                                            float& best, int& arg) {
    float ve = sv[0] + tc[0]; int ie = 0;
    float vo = sv[1] + tc[1]; int io = 1;
#pragma unroll
    for (int i = 2; i < C; i += 2) {
        float v = sv[i] + tc[i];
        if (v > ve) { ve = v; ie = i; }
    }
#pragma unroll
    for (int i = 3; i < C; i += 2) {
        float v = sv[i] + tc[i];
        if (v > vo) { vo = v; io = i; }
    }
    best = ve; arg = ie;
    if (vo > ve || (vo == ve && io < ie)) { best = vo; arg = io; }
}

__global__ __launch_bounds__(NTH) void crf_forward(
    const float* __restrict__ x,       // [B][T][C]
    const float* __restrict__ startT,  // [C]
    const float* __restrict__ endT,    // [C]
    const float* __restrict__ trans,   // [C][C]
    int* __restrict__ out,             // [B][T]
    uint8_t* __restrict__ hist,        // [B][T-1][C]
    int* __restrict__ last)            // [B]
{
    __shared__ float sh_trans[C * C];
    __shared__ float sc[2][48];                     // ping-pong score vector
    __shared__ __align__(16) float ebuf[2][TT * C]; // double-buffered emissions
    __shared__ float fin[48];

    const int l  = threadIdx.x;        // 0..31, single wave
    const int b  = blockIdx.x;
    const int j0 = l;                  // first tag column (always < 41)
    const int j1 = l + 32;             // second tag column for lanes 0..8
    const bool has2 = (j1 < C);

    // Stage transitions matrix once (6724 B).
    for (int idx = l; idx < C * C; idx += NTH) sh_trans[idx] = trans[idx];

    // Kick off tile 0 DMA, then wait; single-wave WG: barrier is a NOP,
    // visibility comes from this wave's ASYNCcnt/DScnt.
    const float* xb = x + (size_t)b * TSTEPS * C;
    copy_tile((const float4*)xb, &ebuf[0][0], l);
    async_wait0();
    __syncthreads();

    // Per-lane trans columns in registers.
    float tc0[C], tc1[C];
    float start0, end0, start1 = 0.f, end1 = 0.f;
#pragma unroll
    for (int i = 0; i < C; ++i) tc0[i] = sh_trans[i * C + j0];
    start0 = startT[j0];
    end0   = endT[j0];
    if (has2) {
#pragma unroll
        for (int i = 0; i < C; ++i) tc1[i] = sh_trans[i * C + j1];
        start1 = startT[j1];
        end1   = endT[j1];
    }

    uint8_t* histb = hist + (size_t)b * (TSTEPS - 1) * C;

    int p = 0;
    for (int k = 0; k < NT; ++k) {
        // Issue next tile's DMA into the other buffer (overlaps compute).
        if (k + 1 < NT)
            copy_tile((const float4*)(xb + (size_t)(k + 1) * TT * C),
                      &ebuf[(k + 1) & 1][0], l);

        const float* eb = &ebuf[k & 1][0];
        for (int s = 0; s < TT; ++s) {
            const int t = k * TT + s;
            float e0 = eb[s * C + j0];
            if (t == 0) {
                sc[p][j0] = start0 + e0;            // score0 = start + emit[0]
                if (has2) sc[p][j1] = start1 + eb[s * C + j1];
            } else {
                // Load the 41-float score vector once (vectorized ds loads),
                // reuse for both owned columns.
                float sv[C];
#pragma unroll
                for (int i = 0; i < C; ++i) sv[i] = sc[p][i];

                float best0; int arg0;
                maxplus_col(sv, tc0, best0, arg0);
                sc[p ^ 1][j0] = best0 + e0;
                histb[(size_t)(t - 1) * C + j0] = (uint8_t)arg0;

                if (has2) {
                    float best1; int arg1;
                    maxplus_col(sv, tc1, best1, arg1);
                    sc[p ^ 1][j1] = best1 + eb[s * C + j1];
                    histb[(size_t)(t - 1) * C + j1] = (uint8_t)arg1;
                }
            }
            __syncthreads();   // single-wave WG: S_NOP + DScnt ordering
            if (t != 0) p ^= 1;
        }
        // Next tile's DMA must have landed before we read it.
        async_wait0();
        __syncthreads();
    }

    // Final: add end transitions, argmax over 41 tags.
    fin[j0] = sc[p][j0] + end0;
    if (has2) fin[j1] = sc[p][j1] + end1;
    __syncthreads();
    if (l == 0) {
        float best = fin[0];
        int   arg  = 0;
        for (int i = 1; i < C; ++i)
            if (fin[i] > best) { best = fin[i]; arg = i; }
        out[(size_t)b * TSTEPS + (TSTEPS - 1)] = arg;
        last[b] = arg;
    }
}

__global__ __launch_bounds__(128) void crf_backtrace(
    const uint8_t* __restrict__ hist,  // [B][T-1][C]
    const int* __restrict__ last,      // [B]
    int* __restrict__ out)             // [B][T]
{
    const int b = blockIdx.x * blockDim.x + threadIdx.x;
    if (b >= BATCH) return;

    const uint8_t* hb = hist + (size_t)b * (TSTEPS - 1) * C;
    int tag = last[b];
    for (int t = TSTEPS - 2; t >= 0; --t) {
        const uint8_t* row = hb + (size_t)t * C;
        // Break the dependent-load chain: prefetch 8 rows (328 B) behind.
        if (t >= 8) __builtin_prefetch(row - 8 * C, 0, 0);
        tag = row[tag];
        out[(size_t)b * TSTEPS + t] = tag;
    }
}

extern "C" void kernel_launch(void* const* d_in, const int* in_sizes, int n_in,
                              void* d_out, int out_size, void* d_ws, size_t ws_size,
                              hipStream_t stream) {
    const float* x  = (const float*)d_in[0];  // [B,T,C]
    const float* st = (const float*)d_in[1];  // [C]
    const float* en = (const float*)d_in[2];  // [C]
    const float* tr = (const float*)d_in[3];  // [C,C]
    int* out = (int*)d_out;                   // [B,T] int32

    // Workspace layout: uint8 history [B][T-1][C] (~43 MB), then int32 last[B].
    uint8_t* hist = (uint8_t*)d_ws;
    const size_t HB = (size_t)BATCH * (TSTEPS - 1) * C;  // 42,949,632 B (16-aligned)
    int* last = (int*)((char*)d_ws + HB);

    crf_forward<<<BATCH, NTH, 0, stream>>>(x, st, en, tr, out, hist, last);
    crf_backtrace<<<BATCH / 128, 128, 0, stream>>>(hist, last, out);
}